// HGT_5506148074101
// MI455X (gfx1250) — compile-verified
//
#include <hip/hip_runtime.h>
#include <math.h>

// ---------------------------------------------------------------------------
// HGT forward for MI455X (gfx1250, wave32).
//
// Precision choice: the edge gather/scatter phase (~1.3GB irregular traffic,
// 128M f32 atomics) dominates runtime at 23.3 TB/s HBM, so GEMMs are NOT the
// bottleneck -> keep exact fp32 numerics and use V_WMMA_F32_16X16X4_F32
// (1024 MACs/instr) instead of lossy bf16/fp8 WMMA.
//
// GEMM data path: weight panel (K x 16) staged in LDS once per block and
// shared by all 8 waves; B fragments come from ds_load_b64, A fragments from
// one aligned global_load_b64 per step; K-loop unrolled 4x.
//
// Param leaves (d_in[8..71]) follow jax.tree_util flatten order:
// top keys sorted: g_mlp, layers, proj_in, y_mlp; dict keys sorted
// alphabetically at every level ('W' < 'b' in ASCII).
// ---------------------------------------------------------------------------

typedef __attribute__((ext_vector_type(2))) float v2f;
typedef __attribute__((ext_vector_type(8))) float v8f;

#define HEADS 4
#define HD 32
#define FH 128
#define EPSLN 1e-5f
#define SCALE_D 0.17677669529663687f /* 1/sqrt(32) */
#define KMAX 128                     /* max GEMM depth (LDS panel bound) */

__device__ __forceinline__ float geluf(float x) {
    return 0.5f * x * (1.0f + erff(x * 0.70710678118654752f));
}
// order-preserving float<->uint map for atomicMax on floats (incl. negatives)
__device__ __forceinline__ unsigned fenc(float x) {
    unsigned u = __float_as_uint(x);
    return (u & 0x80000000u) ? ~u : (u | 0x80000000u);
}
__device__ __forceinline__ float fdec(unsigned u) {
    return (u & 0x80000000u) ? __uint_as_float(u & 0x7fffffffu)
                             : __uint_as_float(~u);
}
#define FENC_NEGINF 0x007FFFFFu /* fenc(-inf) */

// ---------------------------------------------------------------------------
// WMMA fp32 GEMM:  C[n,M] = act(A)[n,K] @ W[K,M] + bias, optional gated-skip.
// One wave per 16x16 tile; block = 8 waves sharing one LDS-staged W panel.
// ACTA: 0=identity, 1=exact GELU on A. POST: 0=store, 1=g*val+(1-g)*S.
// Requires: K % 4 == 0, K <= KMAX, lda even (8B-aligned rows).
// ---------------------------------------------------------------------------
template <int ACTA, int POST>
__global__ __launch_bounds__(256) void gemm16_wmma(
    const float* __restrict__ A, int lda, const float* __restrict__ W, int ldw,
    const float* __restrict__ bias, float* __restrict__ C, int ldc,
    const float* __restrict__ S, const float* __restrict__ skipParam,
    int nRows, int K) {
    __shared__ v2f ldsB[(KMAX / 2) * 16];  // pair-interleaved: [k/2][col]

    const int col0 = blockIdx.y * 16;
    // cooperative stage of W panel (cols col0..col0+15, all K rows) into LDS
    for (int idx = threadIdx.x; idx < (K >> 1) * 16; idx += 256) {
        const int pr = idx >> 4, c = idx & 15;
        v2f w;
        w.x = W[(size_t)(2 * pr) * ldw + col0 + c];
        w.y = W[(size_t)(2 * pr + 1) * ldw + col0 + c];
        ldsB[idx] = w;
    }
    __syncthreads();

    const int wave = threadIdx.x >> 5;
    const int lane = threadIdx.x & 31;
    const int row0 = (blockIdx.x * 8 + wave) * 16;
    if (row0 >= nRows) return;   // wave-uniform; after barrier, EXEC stays full
    const int half = lane >> 4;  // lanes 0-15: K+0/K+1, lanes 16-31: K+2/K+3
    const int lr = lane & 15;

    int rowA = row0 + lr;
    if (rowA > nRows - 1) rowA = nRows - 1;  // clamp (keeps EXEC all-ones)
    const float* Arow = A + (size_t)rowA * lda;

    v8f acc = {0.f, 0.f, 0.f, 0.f, 0.f, 0.f, 0.f, 0.f};
#pragma unroll 4
    for (int k = 0; k < K; k += 4) {
        const int kk = k + half * 2;                    // even -> 8B aligned
        v2f av = *(const v2f*)(Arow + kk);              // global_load_b64
        if (ACTA == 1) { av.x = geluf(av.x); av.y = geluf(av.y); }
        const v2f bv = ldsB[((k >> 1) + half) * 16 + lr];  // ds_load_b64
        acc = __builtin_amdgcn_wmma_f32_16x16x4_f32(
            false, av, false, bv, (short)0, acc, false, false);
    }
    float gate = 0.f;
    if (POST == 1) gate = 1.f / (1.f + expf(-skipParam[0]));
#pragma unroll
    for (int r = 0; r < 8; ++r) {
        const int row = row0 + r + half * 8;  // C layout: vgpr r -> M=r / r+8
        if (row >= nRows) continue;
        const int col = col0 + lr;
        float v = acc[r];
        if (bias) v += bias[col];
        if (POST == 1) v = gate * v + (1.f - gate) * S[(size_t)row * ldc + col];
        C[(size_t)row * ldc + col] = v;
    }
}

// ---------------------------------------------------------------------------
// Fold per-head relation matrix into a dense 128x128 weight:
//   Out[r][h*32+e] = sum_d A[r*lda + h*32 + d] * R[h][d][e]
// Used for W (nRows=128, lda=384 with k/v column offset) and bias (nRows=1).
// ---------------------------------------------------------------------------
__global__ void rel_combine(const float* __restrict__ A, int lda,
                            const float* __restrict__ R, float* __restrict__ Out,
                            int nRows) {
    const int tid = blockIdx.x * blockDim.x + threadIdx.x;
    if (tid >= nRows * FH) return;
    const int r = tid >> 7, c = tid & 127, h = c >> 5, e = c & 31;
    const float* a = A + (size_t)r * lda + h * HD;
    const float* rr = R + h * (HD * HD) + e;
    float acc = 0.f;
#pragma unroll
    for (int d = 0; d < HD; ++d) acc += a[d] * rr[d * HD];
    Out[tid] = acc;
}

// ---------------------------------------------------------------------------
__global__ void fill_u32(unsigned* __restrict__ p, unsigned v, size_t n) {
    size_t i = (size_t)blockIdx.x * blockDim.x + threadIdx.x;
    if (i < n) p[i] = v;
}

// graph layer norm stats, wave32 per node
__global__ __launch_bounds__(256) void gln_stats1(
    const float* __restrict__ X, const int* __restrict__ batch,
    float* __restrict__ gsum, float* __restrict__ gcnt, int n) {
    const int gw = (int)((blockIdx.x * (size_t)blockDim.x + threadIdx.x) >> 5);
    const int lane = threadIdx.x & 31;
    if (gw >= n) return;
    const float* xr = X + (size_t)gw * FH;
    float s = xr[lane] + xr[lane + 32] + xr[lane + 64] + xr[lane + 96];
    for (int o = 16; o; o >>= 1) s += __shfl_xor(s, o);
    if (lane == 0) {
        const int g = batch[gw];
        atomicAdd(&gsum[g], s);
        atomicAdd(&gcnt[g], 1.0f);
    }
}
__global__ __launch_bounds__(256) void gln_stats2(
    const float* __restrict__ X, const int* __restrict__ batch,
    const float* __restrict__ gsum, const float* __restrict__ gcnt,
    float* __restrict__ gssq, int n) {
    const int gw = (int)((blockIdx.x * (size_t)blockDim.x + threadIdx.x) >> 5);
    const int lane = threadIdx.x & 31;
    if (gw >= n) return;
    const int g = batch[gw];
    const float cnt = fmaxf(gcnt[g], 1.f) * (float)FH;
    const float mean = gsum[g] / cnt;
    const float* xr = X + (size_t)gw * FH;
    float s = 0.f;
    for (int i = lane; i < FH; i += 32) {
        const float d = xr[i] - mean;
        s += d * d;
    }
    for (int o = 16; o; o >>= 1) s += __shfl_xor(s, o);
    if (lane == 0) atomicAdd(&gssq[g], s);
}
__global__ void gln_norm(const float* __restrict__ X,
                         const int* __restrict__ batch,
                         const float* __restrict__ gsum,
                         const float* __restrict__ gcnt,
                         const float* __restrict__ gssq,
                         const float* __restrict__ w, const float* __restrict__ b,
                         float* __restrict__ XN, size_t total) {
    const size_t i = (size_t)blockIdx.x * blockDim.x + threadIdx.x;
    if (i >= total) return;
    const int node = (int)(i >> 7), f = (int)(i & 127);
    const int g = batch[node];
    const float cnt = fmaxf(gcnt[g], 1.f) * (float)FH;
    const float mean = gsum[g] / cnt;
    const float var = gssq[g] / cnt;
    XN[i] = (X[i] - mean) * rsqrtf(var + EPSLN) * w[f] + b[f];
}

// ---- edge phase ----
__global__ void edge_alpha(const int* __restrict__ eS, const int* __restrict__ eD,
                           int nE, const float* __restrict__ Q,
                           const float* __restrict__ KE,
                           const float* __restrict__ pRel,
                           float* __restrict__ alpha, unsigned* __restrict__ mx) {
    const int tid = blockIdx.x * blockDim.x + threadIdx.x;
    if (tid >= nE * HEADS) return;
    const int e = tid >> 2, h = tid & 3;
    const int s = eS[e], t = eD[e];
    const float* q = Q + (size_t)t * FH + h * HD;
    const float* k = KE + (size_t)s * FH + h * HD;
    float acc = 0.f;
#pragma unroll
    for (int d = 0; d < HD; ++d) acc += q[d] * k[d];
    acc *= pRel[h] * SCALE_D;
    alpha[tid] = acc;
    atomicMax(&mx[(size_t)t * HEADS + h], fenc(acc));
}
__global__ void edge_exp(const int* __restrict__ eD, int nE,
                         const unsigned* __restrict__ mx,
                         float* __restrict__ alpha, float* __restrict__ den) {
    const int tid = blockIdx.x * blockDim.x + threadIdx.x;
    if (tid >= nE * HEADS) return;
    const int e = tid >> 2, h = tid & 3;
    const int t = eD[e];
    const float ea = expf(alpha[tid] - fdec(mx[(size_t)t * HEADS + h]));
    alpha[tid] = ea;
    atomicAdd(&den[(size_t)t * HEADS + h], ea);
}
// one wave per (edge, head): 32 lanes scatter-add one head of the message
__global__ __launch_bounds__(256) void edge_agg(
    const int* __restrict__ eS, const int* __restrict__ eD, int nE,
    const float* __restrict__ alpha, const float* __restrict__ den,
    const float* __restrict__ VE, float* __restrict__ AGG) {
    const long gw = ((long)blockIdx.x * blockDim.x + threadIdx.x) >> 5;
    const int lane = threadIdx.x & 31;
    if (gw >= (long)nE * HEADS) return;
    const int e = (int)(gw >> 2), h = (int)(gw & 3);
    const int s = eS[e], t = eD[e];
    const float w = alpha[gw] / den[(size_t)t * HEADS + h];
    const float v = VE[(size_t)s * FH + h * HD + lane];
    atomicAdd(&AGG[(size_t)t * FH + h * HD + lane], w * v);
}

// ---- pooling ----
__global__ void pool_kernel(const float* __restrict__ X,
                            const int* __restrict__ batch,
                            float* __restrict__ pAdd, unsigned* __restrict__ pMax,
                            size_t total) {
    const size_t i = (size_t)blockIdx.x * blockDim.x + threadIdx.x;
    if (i >= total) return;
    const int node = (int)(i >> 7), f = (int)(i & 127);
    const int g = batch[node];
    const float v = X[i];
    atomicAdd(&pAdd[(size_t)g * FH + f], v);
    atomicMax(&pMax[(size_t)g * FH + f], fenc(v));
}

// ---- final head MLP (tiny: B=64 rows; one thread per row) ----
__global__ void head_kernel(const float* pAdd0, const unsigned* pMax0,
                            const float* pAdd1, const unsigned* pMax1,
                            const float* yBase, const float* y1W, const float* y1b,
                            const float* y2W, const float* y2b, const float* g1W,
                            const float* g1b, const float* ln1w, const float* ln1b,
                            const float* g2W, const float* g2b, const float* ln2w,
                            const float* ln2b, const float* g3W, const float* g3b,
                            float* out, int nB) {
    const int r = blockIdx.x * blockDim.x + threadIdx.x;
    if (r >= nB) return;
    float hv[528];
    for (int i = 0; i < FH; ++i) hv[i] = pAdd0[(size_t)r * FH + i];
    for (int i = 0; i < FH; ++i) {
        float m = fdec(pMax0[(size_t)r * FH + i]);
        hv[FH + i] = (isinf(m) && m < 0.f) ? 0.f : m;
    }
    for (int i = 0; i < FH; ++i) hv[256 + i] = pAdd1[(size_t)r * FH + i];
    for (int i = 0; i < FH; ++i) {
        float m = fdec(pMax1[(size_t)r * FH + i]);
        hv[384 + i] = (isinf(m) && m < 0.f) ? 0.f : m;
    }
    const float yb = yBase[r];
    float t16[16];
    for (int j = 0; j < 16; ++j) {
        const float v = yb * y1W[j] + y1b[j];
        t16[j] = v >= 0.f ? v : 0.1f * v;  // LeakyReLU(0.1)
    }
    for (int j = 0; j < 16; ++j) {
        float a = y2b[j];
        for (int i = 0; i < 16; ++i) a += t16[i] * y2W[i * 16 + j];
        hv[512 + j] = a;
    }
    // l1 528->128, row LN, gelu
    float h1[128];
    for (int j = 0; j < 128; ++j) {
        float a = g1b[j];
        for (int i = 0; i < 528; ++i) a += hv[i] * g1W[i * 128 + j];
        h1[j] = a;
    }
    float mu = 0.f;
    for (int j = 0; j < 128; ++j) mu += h1[j];
    mu /= 128.f;
    float vr = 0.f;
    for (int j = 0; j < 128; ++j) { float d = h1[j] - mu; vr += d * d; }
    vr /= 128.f;
    const float is1 = rsqrtf(vr + EPSLN);
    for (int j = 0; j < 128; ++j)
        h1[j] = geluf((h1[j] - mu) * is1 * ln1w[j] + ln1b[j]);
    // l2 128->64, row LN, gelu
    float h2[64];
    for (int j = 0; j < 64; ++j) {
        float a = g2b[j];
        for (int i = 0; i < 128; ++i) a += h1[i] * g2W[i * 64 + j];
        h2[j] = a;
    }
    mu = 0.f;
    for (int j = 0; j < 64; ++j) mu += h2[j];
    mu /= 64.f;
    vr = 0.f;
    for (int j = 0; j < 64; ++j) { float d = h2[j] - mu; vr += d * d; }
    vr /= 64.f;
    const float is2 = rsqrtf(vr + EPSLN);
    for (int j = 0; j < 64; ++j)
        h2[j] = geluf((h2[j] - mu) * is2 * ln2w[j] + ln2b[j]);
    float o = g3b[0];
    for (int i = 0; i < 64; ++i) o += h2[i] * g3W[i];
    out[r] = o;
}

// ---------------------------------------------------------------------------
static inline unsigned gdiv(size_t n, unsigned b) {
    return (unsigned)((n + b - 1) / b);
}

extern "C" void kernel_launch(void* const* d_in, const int* in_sizes, int n_in,
                              void* d_out, int out_size, void* d_ws,
                              size_t ws_size, hipStream_t stream) {
    const float* x_in[2] = {(const float*)d_in[0], (const float*)d_in[1]};
    const int* e_flow = (const int*)d_in[2];
    const int* e_def = (const int*)d_in[3];
    const int* e_use = (const int*)d_in[4];
    const int* batch[2] = {(const int*)d_in[5], (const int*)d_in[6]};
    const float* y_base = (const float*)d_in[7];
    auto pf = [&](int i) { return (const float*)d_in[i]; };

    const int N = in_sizes[0] / 64;  // CIN = 64
    const int nB = out_size;         // 64 graphs
    const int nE[3] = {in_sizes[2] / 2, in_sizes[3] / 2, in_sizes[4] / 2};
    const int* eS[3] = {e_flow, e_def + 0, e_use};
    const int* eD[3] = {e_flow + nE[0], e_def + nE[1], e_use + nE[2]};
    // rel r: 0=flow(op->op) 1=def(op->var) 2=use(var->op)
    const int relSrc[3] = {0, 0, 1};
    const int relDst[3] = {0, 1, 0};

    const size_t NF = (size_t)N * FH;
    float* p = (float*)d_ws;
    auto alloc = [&](size_t n) { float* r = p; p += n; return r; };
    float* X[2] = {alloc(NF), alloc(NF)};
    float* XN[2] = {alloc(NF), alloc(NF)};
    float* Q[2] = {alloc(NF), alloc(NF)};
    float* KE[3] = {alloc(NF), alloc(NF), alloc(NF)};
    float* VE[3] = {alloc(NF), alloc(NF), alloc(NF)};
    float* AGG[2] = {alloc(NF), alloc(NF)};
    float* alphaB[3] = {alloc((size_t)nE[0] * HEADS), alloc((size_t)nE[1] * HEADS),
                        alloc((size_t)nE[2] * HEADS)};
    unsigned* mx[2] = {(unsigned*)alloc((size_t)N * HEADS),
                       (unsigned*)alloc((size_t)N * HEADS)};
    float* den[2] = {alloc((size_t)N * HEADS), alloc((size_t)N * HEADS)};
    float* Wke[3] = {alloc(FH * FH), alloc(FH * FH), alloc(FH * FH)};
    float* bke[3] = {alloc(FH), alloc(FH), alloc(FH)};
    float* Wve[3] = {alloc(FH * FH), alloc(FH * FH), alloc(FH * FH)};
    float* bve[3] = {alloc(FH), alloc(FH), alloc(FH)};
    float* gstat = alloc(6 * 64);  // [gsum0 gcnt0 gssq0 gsum1 gcnt1 gssq1]x64
    float* pAdd[2] = {alloc((size_t)nB * FH), alloc((size_t)nB * FH)};
    unsigned* pMax[2] = {(unsigned*)alloc((size_t)nB * FH),
                         (unsigned*)alloc((size_t)nB * FH)};

    const unsigned rowBlk = gdiv((size_t)gdiv((size_t)N, 16), 8);

    // ---- proj_in: X[t] = x_in[t] @ W(64x128) + b ----
    for (int t = 0; t < 2; ++t)
        gemm16_wmma<0, 0><<<dim3(rowBlk, FH / 16), 256, 0, stream>>>(
            x_in[t], 64, pf(64 + 2 * t), FH, pf(65 + 2 * t), X[t], FH, nullptr,
            nullptr, N, 64);

    for (int l = 0; l < 2; ++l) {
        const int Lb = 18 + l * 23;  // first leaf of this layer (sorted order)
        const float* k_rel[3] = {pf(Lb + 1), pf(Lb + 0), pf(Lb + 2)};
        const float* v_rel[3] = {pf(Lb + 21), pf(Lb + 20), pf(Lb + 22)};
        const float* p_rel[3] = {pf(Lb + 16), pf(Lb + 15), pf(Lb + 17)};
        const float* kqvW[2] = {pf(Lb + 3), pf(Lb + 5)};
        const float* kqvB[2] = {pf(Lb + 4), pf(Lb + 6)};
        const float* nrmB[2] = {pf(Lb + 7), pf(Lb + 9)};
        const float* nrmW[2] = {pf(Lb + 8), pf(Lb + 10)};
        const float* outW[2] = {pf(Lb + 11), pf(Lb + 13)};
        const float* outB[2] = {pf(Lb + 12), pf(Lb + 14)};
        const float* skip[2] = {pf(Lb + 18), pf(Lb + 19)};

        // graph layer norm
        fill_u32<<<gdiv(6 * 64, 256), 256, 0, stream>>>((unsigned*)gstat, 0u,
                                                        6 * 64);
        for (int t = 0; t < 2; ++t) {
            float* gs = gstat + t * 3 * 64;
            gln_stats1<<<gdiv((size_t)N * 32, 256), 256, 0, stream>>>(
                X[t], batch[t], gs, gs + 64, N);
            gln_stats2<<<gdiv((size_t)N * 32, 256), 256, 0, stream>>>(
                X[t], batch[t], gs, gs + 64, gs + 128, N);
            gln_norm<<<gdiv(NF, 256), 256, 0, stream>>>(
                X[t], batch[t], gs, gs + 64, gs + 128, nrmW[t], nrmB[t], XN[t],
                NF);
        }

        // fold relation matrices into dense 128x128 weights (+bias)
        for (int r = 0; r < 3; ++r) {
            const int s = relSrc[r];
            rel_combine<<<gdiv(FH * FH, 256), 256, 0, stream>>>(
                kqvW[s], 3 * FH, k_rel[r], Wke[r], FH);
            rel_combine<<<1, FH, 0, stream>>>(kqvB[s], 0, k_rel[r], bke[r], 1);
            rel_combine<<<gdiv(FH * FH, 256), 256, 0, stream>>>(
                kqvW[s] + 2 * FH, 3 * FH, v_rel[r], Wve[r], FH);
            rel_combine<<<1, FH, 0, stream>>>(kqvB[s] + 2 * FH, 0, v_rel[r],
                                              bve[r], 1);
        }

        // q per type; ke/ve per relation (all WMMA GEMMs, K=128, M=128)
        for (int t = 0; t < 2; ++t)
            gemm16_wmma<0, 0><<<dim3(rowBlk, FH / 16), 256, 0, stream>>>(
                XN[t], FH, kqvW[t] + FH, 3 * FH, kqvB[t] + FH, Q[t], FH, nullptr,
                nullptr, N, FH);
        for (int r = 0; r < 3; ++r) {
            const int s = relSrc[r];
            gemm16_wmma<0, 0><<<dim3(rowBlk, FH / 16), 256, 0, stream>>>(
                XN[s], FH, Wke[r], FH, bke[r], KE[r], FH, nullptr, nullptr, N,
                FH);
            gemm16_wmma<0, 0><<<dim3(rowBlk, FH / 16), 256, 0, stream>>>(
                XN[s], FH, Wve[r], FH, bve[r], VE[r], FH, nullptr, nullptr, N,
                FH);
        }

        // init softmax stats + aggregation buffers
        for (int t = 0; t < 2; ++t) {
            fill_u32<<<gdiv((size_t)N * HEADS, 256), 256, 0, stream>>>(
                mx[t], FENC_NEGINF, (size_t)N * HEADS);
            fill_u32<<<gdiv((size_t)N * HEADS, 256), 256, 0, stream>>>(
                (unsigned*)den[t], 0u, (size_t)N * HEADS);
            fill_u32<<<gdiv(NF, 256), 256, 0, stream>>>((unsigned*)AGG[t], 0u,
                                                        NF);
        }
        // segment softmax across ALL relations per destination node
        for (int r = 0; r < 3; ++r)
            edge_alpha<<<gdiv((size_t)nE[r] * HEADS, 256), 256, 0, stream>>>(
                eS[r], eD[r], nE[r], Q[relDst[r]], KE[r], p_rel[r], alphaB[r],
                mx[relDst[r]]);
        for (int r = 0; r < 3; ++r)
            edge_exp<<<gdiv((size_t)nE[r] * HEADS, 256), 256, 0, stream>>>(
                eD[r], nE[r], mx[relDst[r]], alphaB[r], den[relDst[r]]);
        for (int r = 0; r < 3; ++r)
            edge_agg<<<gdiv((size_t)nE[r] * HEADS * 32, 256), 256, 0, stream>>>(
                eS[r], eD[r], nE[r], alphaB[r], den[relDst[r]], VE[r],
                AGG[relDst[r]]);

        // out projection: X[t] = g*(gelu(AGG)@Wout+b) + (1-g)*XN   (fused WMMA)
        for (int t = 0; t < 2; ++t)
            gemm16_wmma<1, 1><<<dim3(rowBlk, FH / 16), 256, 0, stream>>>(
                AGG[t], FH, outW[t], FH, outB[t], X[t], FH, XN[t], skip[t], N,
                FH);
    }

    // ---- pooling + head ----
    for (int t = 0; t < 2; ++t) {
        fill_u32<<<gdiv((size_t)nB * FH, 256), 256, 0, stream>>>(
            (unsigned*)pAdd[t], 0u, (size_t)nB * FH);
        fill_u32<<<gdiv((size_t)nB * FH, 256), 256, 0, stream>>>(
            pMax[t], FENC_NEGINF, (size_t)nB * FH);
        pool_kernel<<<gdiv(NF, 256), 256, 0, stream>>>(X[t], batch[t], pAdd[t],
                                                       pMax[t], NF);
    }
    head_kernel<<<1, 64, 0, stream>>>(
        pAdd[0], pMax[0], pAdd[1], pMax[1], y_base, pf(68), pf(69), pf(70),
        pf(71), pf(8), pf(9), pf(15), pf(14), pf(10), pf(11), pf(17), pf(16),
        pf(12), pf(13), (float*)d_out, nB);
}